// PredHead_69672959476066
// MI455X (gfx1250) — compile-verified
//
#include <hip/hip_runtime.h>
#include <math.h>

typedef float v2f __attribute__((ext_vector_type(2)));
typedef float v8f __attribute__((ext_vector_type(8)));

#define B_    8
#define L_    256
#define DP_   128
#define DS_   256
#define NEAR_ 4

__device__ __forceinline__ float wave_sum(float v) {
  v += __shfl_xor(v, 1);
  v += __shfl_xor(v, 2);
  v += __shfl_xor(v, 4);
  v += __shfl_xor(v, 8);
  v += __shfl_xor(v, 16);
  return v;
}

// ---------------------------------------------------------------------------
// LayerNorm over rows of length 256 (one wave per row, 8 rows per block)
// ---------------------------------------------------------------------------
__global__ void ln_rows_kernel(const float* __restrict__ X,
                               const float* __restrict__ g,
                               const float* __restrict__ bt,
                               float* __restrict__ Y) {
  int wave = threadIdx.x >> 5, lane = threadIdx.x & 31;
  int r = blockIdx.x * 8 + wave;
  const float* x = X + (size_t)r * DS_;
  float v[8]; float s = 0.f, sq = 0.f;
#pragma unroll
  for (int u = 0; u < 8; ++u) { float t = x[lane + 32 * u]; v[u] = t; s += t; sq += t * t; }
  s = wave_sum(s); sq = wave_sum(sq);
  float mu = s * (1.f / DS_);
  float rstd = rsqrtf(sq * (1.f / DS_) - mu * mu + 1e-5f);
  float* y = Y + (size_t)r * DS_;
#pragma unroll
  for (int u = 0; u < 8; ++u) { int c = lane + 32 * u; y[c] = (v[u] - mu) * rstd * g[c] + bt[c]; }
}

// ---------------------------------------------------------------------------
// FP32 WMMA GEMM: C[M,N] = A[M,K] * W[K,N]  (one wave per 16x16 tile)
// Uses V_WMMA_F32_16X16X4_F32; K stepped by 4.
// ---------------------------------------------------------------------------
__global__ void gemm_wmma_f32(const float* __restrict__ A,
                              const float* __restrict__ W,
                              float* __restrict__ C,
                              int M, int N, int K) {
  int wave = threadIdx.x >> 5;
  int lane = threadIdx.x & 31;
  int tile = blockIdx.x * 8 + wave;
  int ntn = N >> 4;
  int mt = tile / ntn, nt = tile - mt * ntn;
  if (mt * 16 >= M) return;                 // wave-uniform guard (EXEC stays full)
  int m = mt * 16 + (lane & 15);
  int n = nt * 16 + (lane & 15);
  int khi = (lane & 16) >> 3;               // lanes 16-31 hold K+2/K+3
  v8f acc = {};
  for (int k0 = 0; k0 < K; k0 += 4) {
    int kk = k0 + khi;
    v2f a, b;
    a.x = A[(size_t)m * K + kk];
    a.y = A[(size_t)m * K + kk + 1];
    b.x = W[(size_t)kk * N + n];
    b.y = W[(size_t)(kk + 1) * N + n];
    acc = __builtin_amdgcn_wmma_f32_16x16x4_f32(false, a, false, b, (short)0, acc,
                                                false, false);
  }
  int rbase = mt * 16 + ((lane & 16) >> 1); // hi lanes: rows M=r+8
#pragma unroll
  for (int r = 0; r < 8; ++r)
    C[(size_t)(rbase + r) * N + n] = acc[r];
}

// ---------------------------------------------------------------------------
// Seq branch reduction: masked mean/max over L per channel -> s_feat[B,512]
// ---------------------------------------------------------------------------
__global__ void s_reduce_kernel(const float* __restrict__ S,
                                const unsigned char* __restrict__ mask,
                                float* __restrict__ s_feat) {
  int b = blockIdx.x, c = threadIdx.x;
  float s = 0.f, mx = -INFINITY, n = 0.f;
  for (int i = 0; i < L_; ++i) {
    float v = S[((size_t)(b * L_ + i)) * DS_ + c];
    if (mask[b * L_ + i]) { s += v; mx = fmaxf(mx, v); n += 1.f; }
  }
  s_feat[b * 2 * DS_ + c] = s / fmaxf(n, 1e-6f);
  s_feat[b * 2 * DS_ + DS_ + c] = (n > 0.f) ? mx : 0.f;
}

// ---------------------------------------------------------------------------
// Big streaming kernel: per (b,i) compute LN-space numerators/counts for the
// inter-chain row reduction and the intra near-band reduction.
//   Sum_j f_j * LN(x_j)[c] = g[c]*(S1[c] - A) + beta[c]*n
// with S1[c]=Sum f_j x_jc rstd_j, A=Sum f_j mu_j rstd_j, n=Sum f_j.
// ---------------------------------------------------------------------------
__global__ void pair_reduce_kernel(const float* __restrict__ pair,
                                   const int* __restrict__ chain,
                                   const unsigned char* __restrict__ mask,
                                   const float* __restrict__ gz,
                                   const float* __restrict__ bz,
                                   float* __restrict__ row_numer,
                                   float* __restrict__ row_n,
                                   float* __restrict__ intra_numer,
                                   float* __restrict__ intra_n) {
  __shared__ float muL[L_], rsL[L_], fInt[L_], fBan[L_];
  int b = blockIdx.x >> 8;
  int i = blockIdx.x & 255;
  int tid = threadIdx.x, wave = tid >> 5, lane = tid & 31;
  const float* pb = pair + (size_t)b * L_ * L_ * DP_;

  { // mask / chain flags, one thread per j
    int j = tid;
    bool pm = (mask[b * L_ + i] != 0) && (mask[b * L_ + j] != 0);
    bool same = chain[b * L_ + i] == chain[b * L_ + j];
    int dd = i > j ? i - j : j - i;
    fInt[j] = (pm && !same) ? 1.f : 0.f;
    fBan[j] = (pm && same && dd >= 1 && dd <= NEAR_) ? 1.f : 0.f;
  }

  // pass 1: per-j LN stats (one wave per j, 32 j's per wave)
  for (int t = 0; t < 32; ++t) {
    int j = wave * 32 + t;
    float s = 0.f, sq = 0.f;
#pragma unroll
    for (int u = 0; u < 8; ++u) {
      int c = lane + 32 * u;
      float x = (c < DP_) ? pb[((size_t)i * L_ + j) * DP_ + c]
                          : pb[((size_t)j * L_ + i) * DP_ + (c - DP_)];
      if (j == i) x = 0.f;  // diagonal zeroed before LN (never used anyway)
      s += x; sq += x * x;
    }
    s = wave_sum(s); sq = wave_sum(sq);
    if (lane == 0) {
      float mu = s * (1.f / DS_);
      muL[j] = mu;
      rsL[j] = rsqrtf(sq * (1.f / DS_) - mu * mu + 1e-5f);
    }
  }
  __syncthreads();

  // per-thread (deterministic) scalar sums over LDS
  float A1 = 0.f, n1 = 0.f, A2 = 0.f, n2 = 0.f;
  for (int j = 0; j < L_; ++j) {
    float mr = muL[j] * rsL[j];
    A1 += fInt[j] * mr; n1 += fInt[j];
    A2 += fBan[j] * mr; n2 += fBan[j];
  }

  // pass 2: one thread per channel, accumulate over j (L2-resident re-read)
  int c = tid;
  float s1 = 0.f, s2 = 0.f;
  for (int j = 0; j < L_; ++j) {
    float x = (c < DP_) ? pb[((size_t)i * L_ + j) * DP_ + c]
                        : pb[((size_t)j * L_ + i) * DP_ + (c - DP_)];
    float xr = x * rsL[j];
    s1 += fInt[j] * xr;
    s2 += fBan[j] * xr;
  }
  size_t row = (size_t)(b * L_ + i);
  row_numer[row * DS_ + c]   = gz[c] * (s1 - A1) + bz[c] * n1;
  intra_numer[row * DS_ + c] = gz[c] * (s2 - A2) + bz[c] * n2;
  if (tid == 0) { row_n[row] = n1; intra_n[row] = n2; }
}

// ---------------------------------------------------------------------------
// Build A2 rows 0..2047 (row means) and zero pad rows 2064..2079
// ---------------------------------------------------------------------------
__global__ void prep_a2_div_kernel(const float* __restrict__ row_numer,
                                   const float* __restrict__ row_n,
                                   float* __restrict__ A2) {
  int wave = threadIdx.x >> 5, lane = threadIdx.x & 31;
  int r = blockIdx.x * 8 + wave;
  if (r >= 2080) return;
  if (r < 2048) {
    float inv = 1.f / fmaxf(row_n[r], 1e-6f);
#pragma unroll
    for (int u = 0; u < 8; ++u) {
      int c = lane + 32 * u;
      A2[(size_t)r * DS_ + c] = row_numer[(size_t)r * DS_ + c] * inv;
    }
  } else if (r >= 2064) {
#pragma unroll
    for (int u = 0; u < 8; ++u) A2[(size_t)r * DS_ + lane + 32 * u] = 0.f;
  }
}

// A2 rows 2048+b (global inter mean) and 2056+b (intra-band mean), per batch
__global__ void prep_a2_batch_kernel(const float* __restrict__ row_numer,
                                     const float* __restrict__ row_n,
                                     const float* __restrict__ intra_numer,
                                     const float* __restrict__ intra_n,
                                     float* __restrict__ A2) {
  int b = blockIdx.x, c = threadIdx.x;
  float s1 = 0.f, s2 = 0.f, n1 = 0.f, n2 = 0.f;
  for (int i = 0; i < L_; ++i) {
    size_t r = (size_t)(b * L_ + i);
    s1 += row_numer[r * DS_ + c];
    s2 += intra_numer[r * DS_ + c];
    n1 += row_n[r];
    n2 += intra_n[r];
  }
  A2[(size_t)(2048 + b) * DS_ + c] = s1 / fmaxf(n1, 1e-6f);
  A2[(size_t)(2056 + b) * DS_ + c] = s2 / fmaxf(n2, 1e-6f);
}

// ---------------------------------------------------------------------------
// Scores -> top-3 (stable, jax tie-break) -> inter_feat / intra_feat(lo half)
// ---------------------------------------------------------------------------
__global__ void score_topk_kernel(const float* __restrict__ RM,
                                  const float* __restrict__ row_n,
                                  float* __restrict__ inter_feat,
                                  float* __restrict__ intra_feat) {
  __shared__ float sc[L_];
  __shared__ int topi[3];
  int b = blockIdx.x, tid = threadIdx.x, wave = tid >> 5, lane = tid & 31;
  for (int t = 0; t < 32; ++t) {
    int i = wave * 32 + t;
    const float* r = RM + ((size_t)(b * L_ + i)) * DS_;
    float s = 0.f;
#pragma unroll
    for (int u = 0; u < 8; ++u) { float v = r[lane + 32 * u]; s += v * v; }
    s = wave_sum(s);
    if (lane == 0) sc[i] = (row_n[b * L_ + i] > 0.f) ? sqrtf(s) : -1e9f;
  }
  __syncthreads();
  if (tid == 0) {
    for (int k = 0; k < 3; ++k) {
      float best = -INFINITY; int bi = 0;
      for (int i = 0; i < L_; ++i) {
        bool used = false;
        for (int q = 0; q < k; ++q) used |= (topi[q] == i);
        if (!used && sc[i] > best) { best = sc[i]; bi = i; }
      }
      topi[k] = bi;
    }
  }
  __syncthreads();
  int c = tid;
  float pk = (RM[((size_t)(b * L_ + topi[0])) * DS_ + c] +
              RM[((size_t)(b * L_ + topi[1])) * DS_ + c] +
              RM[((size_t)(b * L_ + topi[2])) * DS_ + c]) * (1.f / 3.f);
  inter_feat[b * 2 * DS_ + c]       = RM[((size_t)(2048 + b)) * DS_ + c];
  inter_feat[b * 2 * DS_ + DS_ + c] = pk;
  intra_feat[b * 2 * DS_ + c]       = RM[((size_t)(2056 + b)) * DS_ + c];
}

// ---------------------------------------------------------------------------
// Band kernel: 16 band pairs/block; LN -> LDS; WMMA(LDS A, W_pair) -> fold
// per-column max -> bmax[block, 256]
// ---------------------------------------------------------------------------
__global__ void band_max_kernel(const float* __restrict__ pair,
                                const int* __restrict__ chain,
                                const unsigned char* __restrict__ mask,
                                const float* __restrict__ gz,
                                const float* __restrict__ bz,
                                const float* __restrict__ Wp,
                                float* __restrict__ bmax) {
  __shared__ float lnA[16 * DS_];
  __shared__ float validS[16];
  int blk = blockIdx.x;
  int b = blk >> 7, grp = blk & 127;
  int tid = threadIdx.x, wave = tid >> 5, lane = tid & 31;
  const float* pb = pair + (size_t)b * L_ * L_ * DP_;

  for (int s = wave * 2; s < wave * 2 + 2; ++s) {
    int g = grp * 16 + s;
    int i = g >> 3, d = g & 7;
    int delta = (d < 4) ? d - 4 : d - 3;   // {-4..-1, 1..4}
    int j = i + delta;
    bool ok = (j >= 0 && j < L_);
    if (ok)
      ok = (mask[b * L_ + i] != 0) && (mask[b * L_ + j] != 0) &&
           (chain[b * L_ + i] == chain[b * L_ + j]);
    float x[8]; float s0 = 0.f, sq = 0.f;
#pragma unroll
    for (int u = 0; u < 8; ++u) {
      int c = lane + 32 * u;
      float v = 0.f;
      if (ok)
        v = (c < DP_) ? pb[((size_t)i * L_ + j) * DP_ + c]
                      : pb[((size_t)j * L_ + i) * DP_ + (c - DP_)];
      x[u] = v; s0 += v; sq += v * v;
    }
    s0 = wave_sum(s0); sq = wave_sum(sq);
    float mu = s0 * (1.f / DS_);
    float rstd = rsqrtf(sq * (1.f / DS_) - mu * mu + 1e-5f);
#pragma unroll
    for (int u = 0; u < 8; ++u) {
      int c = lane + 32 * u;
      lnA[s * DS_ + c] = ok ? ((x[u] - mu) * rstd * gz[c] + bz[c]) : 0.f;
    }
    if (lane == 0) validS[s] = ok ? 1.f : 0.f;
  }
  __syncthreads();

  int khi = (lane & 16) >> 3;
  int arow = lane & 15;
  for (int tc = wave * 2; tc < wave * 2 + 2; ++tc) {  // each wave owns 2 col tiles
    int n = tc * 16 + (lane & 15);
    v8f acc = {};
    for (int k0 = 0; k0 < DS_; k0 += 4) {
      int kk = k0 + khi;
      v2f a, bb;
      a.x = lnA[arow * DS_ + kk];
      a.y = lnA[arow * DS_ + kk + 1];
      bb.x = Wp[(size_t)kk * DS_ + n];
      bb.y = Wp[(size_t)(kk + 1) * DS_ + n];
      acc = __builtin_amdgcn_wmma_f32_16x16x4_f32(false, a, false, bb, (short)0, acc,
                                                  false, false);
    }
    int sbase = (lane & 16) >> 1;   // hi lanes see rows 8..15
    float m = -INFINITY;
#pragma unroll
    for (int r = 0; r < 8; ++r)
      if (validS[sbase + r] > 0.f) m = fmaxf(m, acc[r]);
    m = fmaxf(m, __shfl_xor(m, 16));
    if (lane < 16) bmax[(size_t)blk * DS_ + n] = m;
  }
}

__global__ void band_reduce_kernel(const float* __restrict__ bmax,
                                   float* __restrict__ intra_feat) {
  int b = blockIdx.x, c = threadIdx.x;
  float m = -INFINITY;
  for (int k = 0; k < 128; ++k)
    m = fmaxf(m, bmax[((size_t)(b * 128 + k)) * DS_ + c]);
  intra_feat[b * 2 * DS_ + DS_ + c] = isinf(m) ? 0.f : m;
}

// ---------------------------------------------------------------------------
// Final MLP head (tiny)
// ---------------------------------------------------------------------------
__global__ void head_mlp_kernel(const float* __restrict__ s_feat,
                                const float* __restrict__ inter_feat,
                                const float* __restrict__ intra_feat,
                                const float* w1a, const float* b1a, const float* w2a, const float* b2a,
                                const float* w1b, const float* b1b, const float* w2b, const float* b2b,
                                const float* w1c, const float* b1c, const float* w2c, const float* b2c,
                                const float* fw1, const float* fb1, const float* fw2, const float* fb2,
                                float* __restrict__ out) {
  __shared__ float part[64];
  __shared__ float logits[3];
  int b = blockIdx.x, t = threadIdx.x;
  const float* feats[3] = { s_feat + b * 512, inter_feat + b * 512, intra_feat + b * 512 };
  const float* W1[3] = { w1a, w1b, w1c };
  const float* B1[3] = { b1a, b1b, b1c };
  const float* W2[3] = { w2a, w2b, w2c };
  const float* B2[3] = { b2a, b2b, b2c };
  for (int m = 0; m < 3; ++m) {
    float h = B1[m][t];
    const float* f = feats[m];
    const float* w = W1[m];
    for (int k = 0; k < 512; ++k) h += f[k] * w[k * 64 + t];
    h = fmaxf(h, 0.f);
    part[t] = h * W2[m][t];
    __syncthreads();
    if (t == 0) {
      float s = 0.f;
      for (int q = 0; q < 64; ++q) s += part[q];
      logits[m] = s + B2[m][0];
    }
    __syncthreads();
  }
  if (t == 0) {
    float o = fb2[0];
    for (int q = 0; q < 8; ++q) {
      float h = fb1[q];
      for (int m = 0; m < 3; ++m) h += logits[m] * fw1[m * 8 + q];
      o += fmaxf(h, 0.f) * fw2[q];
    }
    out[b] = o;
  }
}

// ---------------------------------------------------------------------------
extern "C" void kernel_launch(void* const* d_in, const int* in_sizes, int n_in,
                              void* d_out, int out_size, void* d_ws, size_t ws_size,
                              hipStream_t stream) {
  const float* seq   = (const float*)d_in[0];
  const float* pair  = (const float*)d_in[1];
  const int*   chain = (const int*)d_in[2];
  const unsigned char* mask = (const unsigned char*)d_in[3];
  const float* ln_s_g = (const float*)d_in[4];
  const float* ln_s_b = (const float*)d_in[5];
  const float* ln_z_g = (const float*)d_in[6];
  const float* ln_z_b = (const float*)d_in[7];
  const float* W_seq  = (const float*)d_in[8];
  const float* W_pair = (const float*)d_in[9];
  float* out = (float*)d_out;

  float* ws = (float*)d_ws;
  float* lnseq     = ws;                 // 2048*256
  float* s_out     = lnseq + 524288;     // 2048*256
  float* row_n     = s_out + 524288;     // 2048
  float* intra_n   = row_n + 2048;       // 2048
  float* row_numer = intra_n + 2048;     // 2048*256
  float* intra_num = row_numer + 524288; // 2048*256
  float* A2        = intra_num + 524288; // 2080*256
  float* RM        = A2 + 532480;        // 2080*256
  float* bmax      = RM + 532480;        // 1024*256
  float* s_feat    = bmax + 262144;      // 8*512
  float* inter_f   = s_feat + 4096;      // 8*512
  float* intra_f   = inter_f + 4096;     // 8*512

  // --- seq branch ---
  ln_rows_kernel<<<256, 256, 0, stream>>>(seq, ln_s_g, ln_s_b, lnseq);
  gemm_wmma_f32<<<256, 256, 0, stream>>>(lnseq, W_seq, s_out, 2048, 256, 256);
  s_reduce_kernel<<<8, 256, 0, stream>>>(s_out, mask, s_feat);

  // --- pair LN-space reductions (bandwidth-dominant) ---
  pair_reduce_kernel<<<2048, 256, 0, stream>>>(pair, chain, mask, ln_z_g, ln_z_b,
                                               row_numer, row_n, intra_num, intra_n);

  // --- assemble A2 and run the collapsed GEMM ---
  prep_a2_div_kernel<<<260, 256, 0, stream>>>(row_numer, row_n, A2);
  prep_a2_batch_kernel<<<8, 256, 0, stream>>>(row_numer, row_n, intra_num, intra_n, A2);
  gemm_wmma_f32<<<260, 256, 0, stream>>>(A2, W_pair, RM, 2080, 256, 256);

  // --- top-k / inter features ---
  score_topk_kernel<<<8, 256, 0, stream>>>(RM, row_n, inter_f, intra_f);

  // --- intra band max (true per-pair z via WMMA) ---
  band_max_kernel<<<1024, 256, 0, stream>>>(pair, chain, mask, ln_z_g, ln_z_b, W_pair, bmax);
  band_reduce_kernel<<<8, 256, 0, stream>>>(bmax, intra_f);

  // --- head ---
  head_mlp_kernel<<<8, 64, 0, stream>>>(
      s_feat, inter_f, intra_f,
      (const float*)d_in[10], (const float*)d_in[11], (const float*)d_in[12], (const float*)d_in[13],
      (const float*)d_in[14], (const float*)d_in[15], (const float*)d_in[16], (const float*)d_in[17],
      (const float*)d_in[18], (const float*)d_in[19], (const float*)d_in[20], (const float*)d_in[21],
      (const float*)d_in[22], (const float*)d_in[23], (const float*)d_in[24], (const float*)d_in[25],
      out);
}